// SparseConvLayer_74062416052254
// MI455X (gfx1250) — compile-verified
//
#include <hip/hip_runtime.h>

typedef __attribute__((ext_vector_type(2))) float v2f;
typedef __attribute__((ext_vector_type(8))) float v8f;

// Shapes (from reference):
// x:      (8,16,4,64,32,32) f32      conv_w: (8,64,64,3,3)   conv_b: (8,64)
// dec1_w: (8,8,32,3,3)               dec1_b: (8,8)
// dec2_w: (8,8,8,3)                  dec2_b: (8,8)
// outputs: dec_logits (8,16,8)=1024, node_acts (8,16,64,32,32)=8388608, route (8,16)=128

// ---------------------------------------------------------------------------
// Kernel A: h0 = mean over K=4.  Pure bandwidth: 128 MiB read, 32 MiB write.
// ---------------------------------------------------------------------------
__global__ void mean_k_kernel(const float* __restrict__ x,
                              float* __restrict__ h0, int total4) {
  int vi = blockIdx.x * blockDim.x + threadIdx.x;
  if (vi >= total4) return;
  int nb     = vi >> 14;        // (node*16+b), 0..127
  int within = vi & 16383;
  const float4* xv = (const float4*)x;
  long base = ((long)nb * 4) << 14;     // k stride = 16384 float4
  float4 s  = xv[base + within];
  float4 v1 = xv[base + (1L << 14) + within];
  float4 v2 = xv[base + (2L << 14) + within];
  float4 v3 = xv[base + (3L << 14) + within];
  s.x = (s.x + v1.x + v2.x + v3.x) * 0.25f;
  s.y = (s.y + v1.y + v2.y + v3.y) * 0.25f;
  s.z = (s.z + v1.z + v2.z + v3.z) * 0.25f;
  s.w = (s.w + v1.w + v2.w + v3.w) * 0.25f;
  ((float4*)h0)[vi] = s;
}

// ---------------------------------------------------------------------------
// Kernel W: one-shot weight transpose into pair-interleaved, tap-major layout:
//   wt[node][tap=dy*3+dx][ (cin>>1)*64 + m )*2 + (cin&1) ]
// so kernel B can stage each 64x64 tap slab with coalesced float4 copies and
// read A fragments with single ds_load_b64 ops. 294912 elements (1.2 MiB).
// ---------------------------------------------------------------------------
__global__ void wtrans_kernel(const float* __restrict__ cw,
                              float* __restrict__ wt) {
  int o = blockIdx.x * blockDim.x + threadIdx.x;
  if (o >= 8 * 9 * 4096) return;
  int node = o / 36864;
  int r1   = o - node * 36864;
  int tap  = r1 >> 12;            // dydx
  int q    = r1 & 4095;
  int par  = q & 1;
  int idx  = q >> 1;
  int cp   = idx >> 6;
  int mm   = idx & 63;
  int cin  = cp * 2 + par;
  wt[o] = cw[((node * 64 + mm) * 64 + cin) * 9 + tap];
}

// ---------------------------------------------------------------------------
// Kernel B: Conv2d(64->64, 3x3, SAME) as implicit GEMM on the fp32 WMMA pipe.
// K ordered (dy,dx)-major, cin-minor. Workgroup = 4 waves = one (node,b,y) row.
// Wave w owns Cout tile [16w,16w+16) and both 16-pixel x-halves (A reused).
// LDS is cin-pair interleaved so every fragment element pair is one b64 load:
// 3 ds_load_b64 + 2 v_wmma_f32_16x16x4_f32 per K-step of 4.
// 9 taps x 16 steps x 2 halves = 288 WMMA per wave.
// ---------------------------------------------------------------------------
__global__ void __launch_bounds__(128)
conv_wmma_kernel(const float* __restrict__ h0, const float* __restrict__ wt,
                 const float* __restrict__ cb, float* __restrict__ out) {
  // [cp=cin>>1][row r=0..2][x+1][cin&1] : 32*3*34*2 floats = 26112 B
  __shared__ __align__(16) float s_h0[32 * 3 * 34 * 2];
  // [cp][m][cin&1] for current tap : 32*64*2 floats = 16384 B
  __shared__ __align__(16) float s_w[32 * 64 * 2];

  int wg   = blockIdx.x;                // 8*16*32 = 4096
  int node = wg >> 9;
  int rem  = wg & 511;
  int b    = rem >> 5;
  int y    = rem & 31;
  int tid  = threadIdx.x;
  int wave = tid >> 5, lane = tid & 31;

  // Stage h0 rows y-1..y+1, all 64 cin, x in [-1,33) zero padded (6528/128=51),
  // writing into the pair-interleaved layout.
  const float* h0nb = h0 + ((long)(node * 16 + b) * 64) * 1024;
  for (int t = tid; t < 64 * 3 * 34; t += 128) {
    int cin = t / 102;
    int r   = (t / 34) % 3;
    int xp  = t % 34;
    int yy  = y + r - 1;
    int xx  = xp - 1;
    float v = 0.0f;
    if ((unsigned)yy < 32u && (unsigned)xx < 32u)
      v = h0nb[cin * 1024 + yy * 32 + xx];
    s_h0[(((cin >> 1) * 3 + r) * 34 + xp) * 2 + (cin & 1)] = v;
  }

  v8f acc0 = {};                        // x-half 0 (pixels 0..15)
  v8f acc1 = {};                        // x-half 1 (pixels 16..31)
  int m   = (wave << 4) + (lane & 15);  // Cout row fed into A by this lane
  int kph = lane >> 4;                  // K pair-half: 0 -> cin {0,1}, 1 -> {2,3}
  int n   = lane & 15;
  const float4* wtap_base = (const float4*)(wt + (long)node * 9 * 4096);
  const v2f* swp = (const v2f*)s_w;     // [cp][m]
  const v2f* shp = (const v2f*)s_h0;    // [cp][r][xp]

  for (int tap = 0; tap < 9; ++tap) {
    int dy = tap / 3, dx = tap % 3;
    __syncthreads();
    // Coalesced float4 staging of the 64x64 tap slab (1024 f4 / 128 thr = 8).
    const float4* src = wtap_base + tap * 1024;
    float4* dst = (float4*)s_w;
    for (int t = tid; t < 1024; t += 128) dst[t] = src[t];
    __syncthreads();

#pragma unroll
    for (int cp0 = 0; cp0 < 32; cp0 += 2) {
      v2f a  = swp[(cp0 + kph) * 64 + m];                       // ds_load_b64
      int rb = ((cp0 + kph) * 3 + dy) * 34 + dx + n;
      v2f b0 = shp[rb];                                         // ds_load_b64
      v2f b1 = shp[rb + 16];                                    // ds_load_b64
      acc0 = __builtin_amdgcn_wmma_f32_16x16x4_f32(
          false, a, false, b0, (short)0, acc0, false, false);
      acc1 = __builtin_amdgcn_wmma_f32_16x16x4_f32(
          false, a, false, b1, (short)0, acc1, false, false);
    }
  }

  // Write back with bias. C/D layout: vgpr v holds M = v + 8*(lane>=16), N = lane&15.
  int mhi = (lane >> 4) << 3;
  for (int v = 0; v < 8; ++v) {
    int mm = (wave << 4) + mhi + v;
    float bb = cb[node * 64 + mm];
    long oidx = (((long)(node * 16 + b) * 64 + mm) * 32 + y) * 32;
    out[oidx + n]      = acc0[v] + bb;
    out[oidx + 16 + n] = acc1[v] + bb;
  }
}

// ---------------------------------------------------------------------------
// Kernel C: decision head, algebraically collapsed.
// logits[o] = 898*b2[o] + sum_i sum_k w2[o,i,k] * S_k[i]
//   S_0[i] = P[i] - hd[i,898] - hd[i,899]
//   S_1[i] = P[i] - hd[i,0]   - hd[i,899]
//   S_2[i] = P[i] - hd[i,0]   - hd[i,1]
//   P[i]   = 900*b1[i] + sum_{ci,dy,dx} d1w[i,ci,dy,dx] * T[ci,dy,dx]
//   T[ci,dy,dx] = 30x30 window sum of acts[:, :32] at offset (dy,dx)
// ---------------------------------------------------------------------------
__global__ void __launch_bounds__(256)
decision_kernel(const float* __restrict__ acts,
                const float* __restrict__ d1w, const float* __restrict__ d1b,
                const float* __restrict__ d2w, const float* __restrict__ d2b,
                float* __restrict__ logits_out, float* __restrict__ route_out) {
  int node = blockIdx.x >> 4;
  int b    = blockIdx.x & 15;
  int tid  = threadIdx.x;
  __shared__ float sRW[32][3];
  __shared__ float sT[9];
  __shared__ float sP[8];
  __shared__ float sEdge[8][4];
  __shared__ float sLog[8];

  const float* base = acts + ((long)(node * 16 + b) * 64) * 1024;
  if (tid < 8) sP[tid] = 0.0f;
  __syncthreads();

  for (int ci = 0; ci < 32; ++ci) {
    const float* img = base + ci * 1024;
    if (tid < 32) {
      const float* row = img + tid * 32;
      float tot = 0.0f;
      for (int xx = 0; xx < 32; ++xx) tot += row[xx];
      sRW[tid][0] = tot - row[30] - row[31];
      sRW[tid][1] = tot - row[0]  - row[31];
      sRW[tid][2] = tot - row[0]  - row[1];
    }
    __syncthreads();
    if (tid < 9) {
      int dy = tid / 3, dx = tid % 3;
      float t = 0.0f;
      for (int yy = dy; yy < dy + 30; ++yy) t += sRW[yy][dx];
      sT[tid] = t;
    }
    __syncthreads();
    if (tid < 8) {
      const float* w = d1w + ((node * 8 + tid) * 32 + ci) * 9;
      float a = 0.0f;
      for (int r = 0; r < 9; ++r) a += w[r] * sT[r];
      sP[tid] += a;
    }
    __syncthreads();
  }

  if (tid < 32) {
    int i = tid >> 2, e = tid & 3;
    int yy0 = (e < 2) ? 0 : 29;
    int xx0 = (e == 0) ? 0 : (e == 1 ? 1 : (e == 2 ? 28 : 29));
    float acc = d1b[node * 8 + i];
    const float* w = d1w + (long)(node * 8 + i) * 32 * 9;
    for (int ci = 0; ci < 32; ++ci)
      for (int r = 0; r < 9; ++r)
        acc += base[ci * 1024 + (yy0 + r / 3) * 32 + (xx0 + r % 3)] * w[ci * 9 + r];
    sEdge[i][e] = acc;
  }
  __syncthreads();

  if (tid < 8) {
    int o = tid;
    float L = 898.0f * d2b[node * 8 + o];
    for (int i = 0; i < 8; ++i) {
      float P  = 900.0f * d1b[node * 8 + i] + sP[i];
      float S0 = P - sEdge[i][2] - sEdge[i][3];
      float S1 = P - sEdge[i][0] - sEdge[i][3];
      float S2 = P - sEdge[i][0] - sEdge[i][1];
      const float* w2 = d2w + ((node * 8 + o) * 8 + i) * 3;
      L += w2[0] * S0 + w2[1] * S1 + w2[2] * S2;
    }
    logits_out[(node * 16 + b) * 8 + o] = L;
    sLog[o] = L;
  }
  __syncthreads();
  if (tid == 0) {
    int best = 0; float bv = sLog[0];
    for (int o = 1; o < 8; ++o) if (sLog[o] > bv) { bv = sLog[o]; best = o; }
    route_out[node * 16 + b] = (float)best;   // argmax (first-max tiebreak)
  }
}

// ---------------------------------------------------------------------------
extern "C" void kernel_launch(void* const* d_in, const int* in_sizes, int n_in,
                              void* d_out, int out_size, void* d_ws, size_t ws_size,
                              hipStream_t stream) {
  const float* x   = (const float*)d_in[0];
  const float* cw  = (const float*)d_in[1];
  const float* cb  = (const float*)d_in[2];
  const float* d1w = (const float*)d_in[3];
  const float* d1b = (const float*)d_in[4];
  const float* d2w = (const float*)d_in[5];
  const float* d2b = (const float*)d_in[6];

  float* out    = (float*)d_out;
  float* logits = out;                       // 1024
  float* acts   = out + 1024;                // 8388608
  float* route  = out + 1024 + 8388608;      // 128

  float* h0 = (float*)d_ws;                  // 32 MiB scratch
  float* wt = h0 + 8 * 16 * 64 * 1024;       // +1.2 MiB transposed weights

  // W: pack weights tap-major, pair-interleaved (294912 elems)
  wtrans_kernel<<<1152, 256, 0, stream>>>(cw, wt);
  // A: mean over K (bandwidth bound)
  mean_k_kernel<<<8192, 256, 0, stream>>>(x, h0, 2097152);
  // B: 3x3 SAME conv via fp32 WMMA implicit GEMM (9.7 GFLOP)
  conv_wmma_kernel<<<4096, 128, 0, stream>>>(h0, wt, cb, acts);
  // C: collapsed decision head + argmax routing
  decision_kernel<<<128, 256, 0, stream>>>(acts, d1w, d1b, d2w, d2b, logits, route);
}